// DistanceAwareGATv2_25829933318904
// MI455X (gfx1250) — compile-verified
//
#include <hip/hip_runtime.h>
#include <math.h>

typedef float v2f __attribute__((ext_vector_type(2)));
typedef float v8f __attribute__((ext_vector_type(8)));

#define GAT_HEADS 4
#define NEG_SLOPE 0.2f

// ---------------------------------------------------------------- utilities

__global__ void fill_kernel(float* __restrict__ p, float v, long n) {
  long i = (long)blockIdx.x * blockDim.x + threadIdx.x;
  if (i < n) p[i] = v;
}

__device__ __forceinline__ void atomicMaxF(float* addr, float val) {
  unsigned int* a = (unsigned int*)addr;
  unsigned int old = __hip_atomic_load(a, __ATOMIC_RELAXED, __HIP_MEMORY_SCOPE_AGENT);
  while (__uint_as_float(old) < val) {
    unsigned int assumed = old;
    old = atomicCAS(a, assumed, __float_as_uint(val));
    if (old == assumed) break;
  }
}

__device__ __forceinline__ void atomicAddF(float* addr, float val) {
  __hip_atomic_fetch_add(addr, val, __ATOMIC_RELAXED, __HIP_MEMORY_SCOPE_AGENT);
}

// ---------------------------------------------------------------- WMMA GEMM
// C[M,Nout] = A[M,K] @ W[K,Nout], fp32 via V_WMMA_F32_16X16X4_F32.
// One wave per 16x16 output tile; K stepped by 4.
// Fragment layouts per CDNA5 ISA 7.12.2:
//   A (16x4): lane half h = lane>>4, row m = lane&15; vgpr v holds A[m][2h+v]
//   B (4x16): mirrored: vgpr v holds B[2h+v][n], n = lane&15
//   C/D (16x16): vgpr r holds row m = r + 8h, col n = lane&15
__global__ void gemm_wmma_f32_kernel(const float* __restrict__ A,
                                     const float* __restrict__ W,
                                     float* __restrict__ C,
                                     int M, int K, int Nout) {
  int wave = blockIdx.x * (blockDim.x >> 5) + (threadIdx.x >> 5);
  int tilesN = Nout >> 4;
  int tilesM = M >> 4;  // M must be a multiple of 16 (50000 = 3125*16)
  if (wave >= tilesM * tilesN) return;
  int tm = wave / tilesN, tn = wave - tm * tilesN;
  int m0 = tm << 4, n0 = tn << 4;
  int lane = threadIdx.x & 31;
  int half = lane >> 4, l16 = lane & 15;

  v8f acc = {0.f, 0.f, 0.f, 0.f, 0.f, 0.f, 0.f, 0.f};
  const float* Arow = A + (long)(m0 + l16) * K + 2 * half;
  const float* Wcol = W + n0 + l16;
  for (int k = 0; k < K; k += 4) {
    v2f a = *(const v2f*)(Arow + k);  // 8B aligned: k and 2*half both even
    v2f b;
    b.x = Wcol[(long)(k + 2 * half) * Nout];
    b.y = Wcol[(long)(k + 2 * half + 1) * Nout];
    acc = __builtin_amdgcn_wmma_f32_16x16x4_f32(false, a, false, b,
                                                (short)0, acc, false, false);
  }
  float* Cout = C + n0 + l16;
#pragma unroll
  for (int r = 0; r < 8; ++r) {
    int m = m0 + r + 8 * half;
    Cout[(long)m * Nout] = acc[r];
  }
}

// ------------------------------------------------- pass A: score + seg-max
// One wave per (edge, head). ea recomputed in-register; We/att staged in LDS.
template <int C>
__global__ void edge_score_kernel(const float* __restrict__ xl,
                                  const float* __restrict__ xr,
                                  const float* __restrict__ eattr,  // [E,16]
                                  const float* __restrict__ We,     // [16, 4C]
                                  const float* __restrict__ att,    // [4, C]
                                  const int* __restrict__ ei,       // [2, E]
                                  float* __restrict__ score,        // [E, 4]
                                  float* __restrict__ segmax,       // [N, 4]
                                  int E) {
  constexpr int HC = GAT_HEADS * C;
  __shared__ float sWe[16 * HC];
  __shared__ float sAtt[HC];
  for (int i = threadIdx.x; i < 16 * HC; i += blockDim.x) sWe[i] = We[i];
  for (int i = threadIdx.x; i < HC; i += blockDim.x) sAtt[i] = att[i];
  __syncthreads();

  int wid = blockIdx.x * (blockDim.x >> 5) + (threadIdx.x >> 5);
  int e = wid >> 2, h = wid & 3;
  if (e >= E) return;
  int lane = threadIdx.x & 31;
  int sN = ei[e], dN = ei[E + e];

  const float* eav = eattr + (long)e * 16;
  float part = 0.f;
  for (int c = lane; c < C; c += 32) {
    float ea = 0.f;
#pragma unroll
    for (int k = 0; k < 16; ++k) ea = fmaf(eav[k], sWe[k * HC + h * C + c], ea);
    float v = xl[(long)sN * HC + h * C + c] + xr[(long)dN * HC + h * C + c] + ea;
    v = v > 0.f ? v : NEG_SLOPE * v;             // LeakyReLU
    part = fmaf(v, sAtt[h * C + c], part);
  }
#pragma unroll
  for (int off = 16; off; off >>= 1) part += __shfl_xor(part, off, 32);
  if (lane == 0) {
    score[(long)e * GAT_HEADS + h] = part;
    atomicMaxF(segmax + (long)dN * GAT_HEADS + h, part);
  }
}

// ------------------------------------- pass B: p = exp(score - m), seg-sum
__global__ void softmax_norm_kernel(float* __restrict__ score,        // in: s, out: p
                                    const float* __restrict__ segmax,
                                    float* __restrict__ segsum,
                                    const int* __restrict__ ei, int E) {
  long i = (long)blockIdx.x * blockDim.x + threadIdx.x;
  if (i >= (long)E * GAT_HEADS) return;
  int e = (int)(i >> 2), h = (int)(i & 3);
  int dN = ei[E + e];
  float m = segmax[(long)dN * GAT_HEADS + h];
  float p = expf(score[i] - m);
  score[i] = p;
  atomicAddF(&segsum[(long)dN * GAT_HEADS + h], p);
}

// ------------------------------------------ pass C: acc[dst] += alpha*xl[src]
template <int C>
__global__ void edge_aggregate_kernel(const float* __restrict__ score,   // p
                                      const float* __restrict__ segsum,
                                      const float* __restrict__ xl,
                                      const int* __restrict__ ei,
                                      float* __restrict__ acc, int E) {
  constexpr int HC = GAT_HEADS * C;
  int wid = blockIdx.x * (blockDim.x >> 5) + (threadIdx.x >> 5);
  int e = wid >> 2, h = wid & 3;
  if (e >= E) return;
  int lane = threadIdx.x & 31;
  int sN = ei[e], dN = ei[E + e];
  float p = score[(long)e * GAT_HEADS + h];
  float s = segsum[(long)dN * GAT_HEADS + h];
  float alpha = p / (s + 1e-16f);
  for (int c = lane; c < C; c += 32) {
    float v = alpha * xl[(long)sN * HC + h * C + c];
    atomicAddF(&acc[(long)dN * HC + h * C + c], v);
  }
}

// ----------------------------------------------------- h = elu(acc + b0)
__global__ void elu_bias_kernel(float* __restrict__ a, const float* __restrict__ b,
                                long total, int F) {
  long i = (long)blockIdx.x * blockDim.x + threadIdx.x;
  if (i >= total) return;
  float v = a[i] + b[i % F];
  a[i] = v > 0.f ? v : expm1f(v);
}

// ------------------------------------- out = mean_h(acc1) + b1  (concat=False)
__global__ void mean_bias_kernel(const float* __restrict__ acc,
                                 const float* __restrict__ b,
                                 float* __restrict__ out, int N) {
  long i = (long)blockIdx.x * blockDim.x + threadIdx.x;
  if (i >= (long)N * 64) return;
  int n = (int)(i >> 6), c = (int)(i & 63);
  const float* a = acc + (long)n * 256 + c;
  out[i] = 0.25f * (a[0] + a[64] + a[128] + a[192]) + b[c];
}

// ---------------------------------------------------------------- launcher
extern "C" void kernel_launch(void* const* d_in, const int* in_sizes, int n_in,
                              void* d_out, int out_size, void* d_ws, size_t ws_size,
                              hipStream_t stream) {
  const float* x     = (const float*)d_in[0];
  const int*   ei    = (const int*)d_in[1];   // [2,E] int32
  const float* eattr = (const float*)d_in[2]; // [E,16]
  const float* Wl0   = (const float*)d_in[3];
  const float* Wr0   = (const float*)d_in[4];
  const float* We0   = (const float*)d_in[5];
  const float* att0  = (const float*)d_in[6];
  const float* b0    = (const float*)d_in[7];
  const float* Wl1   = (const float*)d_in[8];
  const float* Wr1   = (const float*)d_in[9];
  const float* We1   = (const float*)d_in[10];
  const float* att1  = (const float*)d_in[11];
  const float* b1    = (const float*)d_in[12];
  float* out = (float*)d_out;

  const int N = in_sizes[0] / 128;   // 50000
  const int E = in_sizes[1] / 2;     // 1600000

  // workspace carve-up (fp32)
  float* w = (float*)d_ws;
  float* xl0     = w; w += (size_t)N * 64;
  float* xr0     = w; w += (size_t)N * 64;
  float* score0  = w; w += (size_t)E * GAT_HEADS;
  float* segmax0 = w; w += (size_t)N * GAT_HEADS;
  float* segsum0 = w; w += (size_t)N * GAT_HEADS;
  float* acc0    = w; w += (size_t)N * 64;    // layer-0 output -> becomes h
  float* xl1     = w; w += (size_t)N * 256;
  float* xr1     = w; w += (size_t)N * 256;
  float* score1  = w; w += (size_t)E * GAT_HEADS;
  float* segmax1 = w; w += (size_t)N * GAT_HEADS;
  float* segsum1 = w; w += (size_t)N * GAT_HEADS;
  float* acc1    = w; w += (size_t)N * 256;

  auto fill = [&](float* p, float v, long n) {
    fill_kernel<<<dim3((unsigned)((n + 255) / 256)), dim3(256), 0, stream>>>(p, v, n);
  };
  const float NEG_INF = -__builtin_inff();
  fill(segmax0, NEG_INF, (long)N * GAT_HEADS);
  fill(segsum0, 0.f,     (long)N * GAT_HEADS);
  fill(acc0,    0.f,     (long)N * 64);
  fill(segmax1, NEG_INF, (long)N * GAT_HEADS);
  fill(segsum1, 0.f,     (long)N * GAT_HEADS);
  fill(acc1,    0.f,     (long)N * 256);

  auto gemm = [&](const float* A, const float* W_, float* C_, int M, int K, int Nout) {
    int tiles = (M >> 4) * (Nout >> 4);
    int blocks = (tiles + 7) / 8;  // 8 waves (256 threads) per block
    gemm_wmma_f32_kernel<<<dim3((unsigned)blocks), dim3(256), 0, stream>>>(A, W_, C_, M, K, Nout);
  };

  const unsigned edgeBlocks = (unsigned)(((long)E * GAT_HEADS + 7) / 8); // wave per (e,h)
  const unsigned ehThreads  = (unsigned)(((long)E * GAT_HEADS + 255) / 256);

  // ---------------- layer 0 (C=16, concat) ----------------
  gemm(x, Wl0, xl0, N, 128, 64);
  gemm(x, Wr0, xr0, N, 128, 64);
  edge_score_kernel<16><<<dim3(edgeBlocks), dim3(256), 0, stream>>>(
      xl0, xr0, eattr, We0, att0, ei, score0, segmax0, E);
  softmax_norm_kernel<<<dim3(ehThreads), dim3(256), 0, stream>>>(
      score0, segmax0, segsum0, ei, E);
  edge_aggregate_kernel<16><<<dim3(edgeBlocks), dim3(256), 0, stream>>>(
      score0, segsum0, xl0, ei, acc0, E);
  elu_bias_kernel<<<dim3((unsigned)(((long)N * 64 + 255) / 256)), dim3(256), 0, stream>>>(
      acc0, b0, (long)N * 64, 64);   // acc0 is now h

  // ---------------- layer 1 (C=64, mean) ----------------
  gemm(acc0, Wl1, xl1, N, 64, 256);
  gemm(acc0, Wr1, xr1, N, 64, 256);
  edge_score_kernel<64><<<dim3(edgeBlocks), dim3(256), 0, stream>>>(
      xl1, xr1, eattr, We1, att1, ei, score1, segmax1, E);
  softmax_norm_kernel<<<dim3(ehThreads), dim3(256), 0, stream>>>(
      score1, segmax1, segsum1, ei, E);
  edge_aggregate_kernel<64><<<dim3(edgeBlocks), dim3(256), 0, stream>>>(
      score1, segsum1, xl1, ei, acc1, E);
  mean_bias_kernel<<<dim3((unsigned)(((long)N * 64 + 255) / 256)), dim3(256), 0, stream>>>(
      acc1, b1, out, N);
}